// GIN_29738353557972
// MI455X (gfx1250) — compile-verified
//
#include <hip/hip_runtime.h>

typedef __attribute__((ext_vector_type(2))) float v2f;
typedef __attribute__((ext_vector_type(8))) float v8f;

#define GN_NODES 50000
#define GN_EDGES 500000
#define GIN_DIM 256
#define GHID 128
#define GOUT_DIM 7
#define GBATCH 65536
#define GBN_EPS 1e-5f

// ---------------- workspace layout (in floats) ----------------
// agg  : [0,      12.8M)   50000*256 (layer1) / 50000*128 (later)
// hcur : [12.8M,  19.2M)   50000*128
// htmp : [19.2M,  25.6M)   50000*128
// z2   : [25.6M,  34.0M)   65536*128
static const size_t F_AGG  = 0;
static const size_t F_HCUR = 12800000;
static const size_t F_HTMP = 19200000;
static const size_t F_Z2   = 25600000;

// ---------------- edge scatter-add (segment_sum) ----------------
__global__ __launch_bounds__(256)
void scatter_add_kernel(const float* __restrict__ h,
                        const int* __restrict__ src,
                        const int* __restrict__ dst,
                        float* __restrict__ agg,
                        int n_edges, int dim)
{
    int vec = dim >> 2; // float4 chunks per edge row
    long long tid = (long long)blockIdx.x * blockDim.x + threadIdx.x;
    long long total = (long long)n_edges * vec;
    if (tid >= total) return;
    int e = (int)(tid / vec);
    int c = (int)(tid % vec) * 4;
    int s = src[e];
    int d = dst[e];
    const float4 v = *(const float4*)(h + (long long)s * dim + c);
    float* out = agg + (long long)d * dim + c;
    atomicAdd(out + 0, v.x);
    atomicAdd(out + 1, v.y);
    atomicAdd(out + 2, v.z);
    atomicAdd(out + 3, v.w);
}

// ---------------- WMMA fp32 GEMM:  Y[M,128] = act(Afrag[M,K] @ W[K,128]+b) -
// FUSE: A-fragment = (1+eps[layer])*A + A2   (fused GIN node update)
// PAIR: A-fragment = concat(A[pairs[idx[m]][0]], A[pairs[idx[m]][1]]) (K=256)
// mode 0: relu only; mode 1: relu -> batchnorm(inference) -> relu
template<int K, int WAVES, bool FUSE, bool PAIR>
__global__ __launch_bounds__(WAVES * 32)
void gemm_wmma_kernel(const float* __restrict__ A,
                      const float* __restrict__ A2,
                      const float* __restrict__ epsp, int layer,
                      const int* __restrict__ pairs,
                      const int* __restrict__ idx,
                      const float* __restrict__ W,
                      const float* __restrict__ bias,
                      float* __restrict__ Y,
                      int M, int mode,
                      const float* __restrict__ bn_g,
                      const float* __restrict__ bn_b,
                      const float* __restrict__ bn_m,
                      const float* __restrict__ bn_v)
{
    const int TN  = 128;            // output width (fixed)
    const int KP  = K / 2;          // k-pairs
    const int LDP = 144;            // v2f per LDS row: 288 dwords, %64==32 -> no conflicts
    __shared__ v2f Wl[KP * LDP];

    int t = threadIdx.x;
    // stage W pre-paired: Wl[kp][n] = (W[2kp][n], W[2kp+1][n])
    for (int i = t; i < KP * TN; i += WAVES * 32) {
        int kp = i >> 7, n = i & 127;
        v2f w;
        w.x = W[(2 * kp) * TN + n];
        w.y = W[(2 * kp + 1) * TN + n];
        Wl[kp * LDP + n] = w;
    }
    __syncthreads();

    int lane = t & 31;
    int wave = t >> 5;
    int lm   = lane & 15;   // row (A) / col (B,C,D) within tile
    int hf   = lane >> 4;   // lane half selects K pair / M offset

    float e = 0.0f;
    if (FUSE) e = 1.0f + epsp[layer];

    int nstrips = (M + 15) >> 4;
    for (int sIdx = blockIdx.x * WAVES + wave; sIdx < nstrips;
         sIdx += (int)gridDim.x * WAVES)
    {
        int m0 = sIdx << 4;
        int row = m0 + lm;
        if (row >= M) row = M - 1;                 // load clamp (M%16==0 here)

        const float* arow  = A  + (long long)row * K;
        const float* arow2 = FUSE ? (A2 + (long long)row * K) : nullptr;
        const float* base0 = nullptr;
        const float* base1 = nullptr;
        if (PAIR) {
            int p  = idx[row];
            int r0 = pairs[2 * p];
            int r1 = pairs[2 * p + 1];
            base0 = A + (long long)r0 * GHID;          // index with ka <  128
            base1 = A + (long long)r1 * GHID - GHID;   // index with ka >= 128
        }

        v8f acc[8] = {};
        for (int kk = 0; kk < K; kk += 4) {
            int ka = kk + 2 * hf;                  // this lane's K pair
            v2f a;
            if (FUSE) {
                v2f hv = *reinterpret_cast<const v2f*>(arow + ka);
                v2f gv = *reinterpret_cast<const v2f*>(arow2 + ka);
                a.x = e * hv.x + gv.x;
                a.y = e * hv.y + gv.y;
            } else if (PAIR) {
                const float* p = (ka < GHID) ? base0 : base1;
                a = *reinterpret_cast<const v2f*>(p + ka);
            } else {
                a = *reinterpret_cast<const v2f*>(arow + ka);
            }
            int kp = (kk >> 1) + hf;
            #pragma unroll
            for (int nt = 0; nt < 8; nt++) {
                v2f b = Wl[kp * LDP + nt * 16 + lm];   // single ds_load_b64
                acc[nt] = __builtin_amdgcn_wmma_f32_16x16x4_f32(
                    false, a, false, b, (short)0, acc[nt], false, false);
            }
        }

        #pragma unroll
        for (int nt = 0; nt < 8; nt++) {
            #pragma unroll
            for (int v = 0; v < 8; v++) {
                int m = m0 + v + 8 * hf;           // C/D layout: VGPR v -> M=v(+8)
                int n = nt * 16 + lm;              // N = lane
                if (m >= M) continue;
                float val = acc[nt][v] + bias[n];
                val = fmaxf(val, 0.0f);
                if (mode == 1) {
                    val = (val - bn_m[n]) * rsqrtf(bn_v[n] + GBN_EPS) * bn_g[n]
                          + bn_b[n];
                    val = fmaxf(val, 0.0f);
                }
                Y[(long long)m * TN + n] = val;
            }
        }
    }
}

// ---------------- final head: out[M,7] = z2[M,128] @ w[128,7] + b ----------
__global__ __launch_bounds__(256)
void out7_kernel(const float* __restrict__ z2,
                 const float* __restrict__ w,
                 const float* __restrict__ b,
                 float* __restrict__ out, int M)
{
    __shared__ float wl[GHID * GOUT_DIM];
    __shared__ float bl[GOUT_DIM];
    for (int i = threadIdx.x; i < GHID * GOUT_DIM; i += blockDim.x) wl[i] = w[i];
    if (threadIdx.x < GOUT_DIM) bl[threadIdx.x] = b[threadIdx.x];
    __syncthreads();

    long long tid = (long long)blockIdx.x * blockDim.x + threadIdx.x;
    if (tid >= (long long)M * GOUT_DIM) return;
    int m = (int)(tid / GOUT_DIM);
    int o = (int)(tid - (long long)m * GOUT_DIM);
    const float* zr = z2 + (long long)m * GHID;
    float acc = bl[o];
    #pragma unroll 4
    for (int k = 0; k < GHID; k++) acc += zr[k] * wl[k * GOUT_DIM + o];
    out[tid] = acc;
}

// ---------------- host-side orchestration ----------------
static inline int cdiv_i(long long a, long long b) { return (int)((a + b - 1) / b); }

extern "C" void kernel_launch(void* const* d_in, const int* in_sizes, int n_in,
                              void* d_out, int out_size, void* d_ws, size_t ws_size,
                              hipStream_t stream)
{
    (void)in_sizes; (void)n_in; (void)out_size; (void)ws_size;

    const float* feat    = (const float*)d_in[0];
    const int*   src     = (const int*)  d_in[1];
    const int*   dst     = (const int*)  d_in[2];
    const int*   pairs   = (const int*)  d_in[3];
    const int*   idx     = (const int*)  d_in[4];
    const float* eps     = (const float*)d_in[5];
    const float* w1_in   = (const float*)d_in[6];
    const float* b1_in   = (const float*)d_in[7];
    const float* w2_in   = (const float*)d_in[8];
    const float* b2_in   = (const float*)d_in[9];
    const float* hw1     = (const float*)d_in[10];
    const float* hb1     = (const float*)d_in[11];
    const float* hw2     = (const float*)d_in[12];
    const float* hb2     = (const float*)d_in[13];
    const float* bn_g    = (const float*)d_in[14];
    const float* bn_b    = (const float*)d_in[15];
    const float* bn_m    = (const float*)d_in[16];
    const float* bn_v    = (const float*)d_in[17];
    const float* lin_w   = (const float*)d_in[18];
    const float* lin_b   = (const float*)d_in[19];
    const float* int_w1  = (const float*)d_in[20];
    const float* int_b1  = (const float*)d_in[21];
    const float* int_w2  = (const float*)d_in[22];
    const float* int_b2  = (const float*)d_in[23];
    float* out = (float*)d_out;

    float* ws   = (float*)d_ws;
    float* agg  = ws + F_AGG;
    float* hcur = ws + F_HCUR;
    float* htmp = ws + F_HTMP;
    float* z2   = ws + F_Z2;

    const int node_strips = (GN_NODES + 15) / 16;   // 3125
    const int node_blocks = cdiv_i(node_strips, 8); // 391
    const int pair_strips = GBATCH / 16;            // 4096
    const int pair_blocks = cdiv_i(pair_strips, 8); // 512

    // ---- Layer 1 (dim 256 -> 128) ----
    hipMemsetAsync(agg, 0, (size_t)GN_NODES * GIN_DIM * sizeof(float), stream);
    scatter_add_kernel<<<cdiv_i((long long)GN_EDGES * (GIN_DIM / 4), 256), 256, 0, stream>>>(
        feat, src, dst, agg, GN_EDGES, GIN_DIM);
    gemm_wmma_kernel<256, 8, true, false><<<node_blocks, 256, 0, stream>>>(
        feat, agg, eps, 0, nullptr, nullptr,
        w1_in, b1_in, htmp, GN_NODES, 0, nullptr, nullptr, nullptr, nullptr);
    gemm_wmma_kernel<128, 8, false, false><<<node_blocks, 256, 0, stream>>>(
        htmp, nullptr, nullptr, 0, nullptr, nullptr,
        w2_in, b2_in, hcur, GN_NODES, 1,
        bn_g + 0 * GHID, bn_b + 0 * GHID, bn_m + 0 * GHID, bn_v + 0 * GHID);

    // ---- Layers 2..3 (dim 128 -> 128) ----
    for (int i = 0; i < 2; i++) {
        hipMemsetAsync(agg, 0, (size_t)GN_NODES * GHID * sizeof(float), stream);
        scatter_add_kernel<<<cdiv_i((long long)GN_EDGES * (GHID / 4), 256), 256, 0, stream>>>(
            hcur, src, dst, agg, GN_EDGES, GHID);
        gemm_wmma_kernel<128, 8, true, false><<<node_blocks, 256, 0, stream>>>(
            hcur, agg, eps, i + 1, nullptr, nullptr,
            hw1 + (size_t)i * GHID * GHID, hb1 + (size_t)i * GHID,
            htmp, GN_NODES, 0, nullptr, nullptr, nullptr, nullptr);
        gemm_wmma_kernel<128, 8, false, false><<<node_blocks, 256, 0, stream>>>(
            htmp, nullptr, nullptr, 0, nullptr, nullptr,
            hw2 + (size_t)i * GHID * GHID, hb2 + (size_t)i * GHID,
            hcur, GN_NODES, 1,
            bn_g + (size_t)(i + 1) * GHID, bn_b + (size_t)(i + 1) * GHID,
            bn_m + (size_t)(i + 1) * GHID, bn_v + (size_t)(i + 1) * GHID);
    }

    // ---- final node linear: htmp = relu(hcur @ lin_w + lin_b) ----
    gemm_wmma_kernel<128, 8, false, false><<<node_blocks, 256, 0, stream>>>(
        hcur, nullptr, nullptr, 0, nullptr, nullptr,
        lin_w, lin_b, htmp, GN_NODES, 0, nullptr, nullptr, nullptr, nullptr);

    // ---- pair head: z2 = relu(concat(h[p0],h[p1]) @ int_w1 + int_b1) ----
    gemm_wmma_kernel<256, 8, false, true><<<pair_blocks, 256, 0, stream>>>(
        htmp, nullptr, nullptr, 0, pairs, idx,
        int_w1, int_b1, z2, GBATCH, 0, nullptr, nullptr, nullptr, nullptr);
    out7_kernel<<<cdiv_i((long long)GBATCH * GOUT_DIM, 256), 256, 0, stream>>>(
        z2, int_w2, int_b2, out, GBATCH);
}